// Mamba3DecodeStep_21577915695235
// MI455X (gfx1250) — compile-verified
//
#include <hip/hip_runtime.h>
#include <hip/hip_bf16.h>

// ---------------------------------------------------------------------------
// Mamba3 decode step for MI455X (gfx1250, wave32).
// Memory-bound workload (~455 MB mandatory traffic, ~19.5us @ 23.3 TB/s):
//   phase1: proj = x @ W_in^T   (W_in 151.5 MB; 4-way M-tile reuse kept inside
//                                one block so HBM sees it ~once)
//   phase2: fused state update  (h 134 MB NT-read + h_new 134 MB NT-write,
//                                b128 vectorized)
//   phase3: y = gated @ W_out^T (W_out 33.5 MB, L2-resident)
// GEMMs use V_WMMA_F32_16X16X4_F32: full f32 precision, and since the layer is
// bandwidth-bound the f32 matrix pipe costs nothing vs f16/fp8 paths.
// ---------------------------------------------------------------------------

typedef __attribute__((ext_vector_type(2))) float v2f;
typedef __attribute__((ext_vector_type(4))) float v4f;
typedef __attribute__((ext_vector_type(8))) float v8f;

#define N_HEADS  32
#define HEAD_DIM 128
#define D_STATE  128
#define D_MODEL  2048
#define D_INNER  4096
#define BATCH    64
#define BC_DIM   (N_HEADS * D_STATE)                               // 4096
#define TH_DIM   (N_HEADS * (D_STATE / 2))                         // 2048
#define PROJ_DIM (2 * D_INNER + 2 * BC_DIM + 2 * N_HEADS + TH_DIM) // 18496

// proj row offsets
#define XOFF   0
#define ZOFF   (D_INNER)
#define BOFF   (2 * D_INNER)
#define COFF   (2 * D_INNER + BC_DIM)
#define DTOFF  (2 * D_INNER + 2 * BC_DIM)
#define LAMOFF (DTOFF + N_HEADS)
#define THOFF  (LAMOFF + N_HEADS)

// d_out layout (flat, reference return order): y, h_new, B, x_proj, theta_new
__device__ __host__ constexpr size_t OFF_Y  = 0;
__device__ __host__ constexpr size_t OFF_H  = (size_t)BATCH * D_MODEL;
__device__ __host__ constexpr size_t OFF_B  = OFF_H + (size_t)BATCH * N_HEADS * D_STATE * HEAD_DIM;
__device__ __host__ constexpr size_t OFF_XP = OFF_B + (size_t)BATCH * N_HEADS * D_STATE;
__device__ __host__ constexpr size_t OFF_TH = OFF_XP + (size_t)BATCH * N_HEADS * HEAD_DIM;

// ---------------------------------------------------------------------------
// C[M,N] = A[M,K] @ B[N,K]^T   (row-major, K contiguous in both operands)
// One wave owns one 16x16 C tile (v8f accumulator).
// Lane mapping follows the CDNA5 16x16x4 f32 WMMA VGPR layout:
//   A (16x4): lane l<16 holds row m=l, k={0,1}; lane 16+l holds k={2,3}
//   B (4x16): lane n<16 holds col n,  k={0,1}; lane 16+n holds k={2,3}
// Tile order: mt varies fastest, so the 4 M-tiles sharing one B-tile live in
// the same block -> each W_in row leaves HBM once, reuse hits WGP$/L2.
// ---------------------------------------------------------------------------
__global__ __launch_bounds__(256) void gemm_abt_wmma_f32(
    const float* __restrict__ A, const float* __restrict__ B,
    float* __restrict__ C, int M, int N, int K)
{
    const int wave   = blockIdx.x * (blockDim.x >> 5) + (threadIdx.x >> 5);
    const int mtiles = M >> 4;
    const int total  = mtiles * (N >> 4);
    if (wave >= total) return;

    const int mt = wave % mtiles;       // fastest: B-tile shared within block
    const int nt = wave / mtiles;
    const int lane = threadIdx.x & 31;
    const int kh   = lane >> 4;         // which K half-pair this lane owns
    const int l    = lane & 15;

    const float* a_ptr = A + (size_t)(mt * 16 + l) * K + 2 * kh;
    const float* b_ptr = B + (size_t)(nt * 16 + l) * K + 2 * kh;

    v8f acc = {};
    for (int k0 = 0; k0 < K; k0 += 32) {
        // stay ~1KB ahead of the streaming reads (global_prefetch_b8)
        __builtin_prefetch(b_ptr + k0 + 256, 0, 3);
        __builtin_prefetch(a_ptr + k0 + 256, 0, 3);
#pragma unroll
        for (int kk = 0; kk < 32; kk += 4) {
            v2f av = *(const v2f*)(a_ptr + k0 + kk);
            v2f bv = *(const v2f*)(b_ptr + k0 + kk);
            acc = __builtin_amdgcn_wmma_f32_16x16x4_f32(
                /*neg_a=*/false, av, /*neg_b=*/false, bv,
                /*c_mod=*/(short)0, acc, /*reuse_a=*/false, /*reuse_b=*/false);
        }
    }

    // C/D layout: VGPR r -> M = r + 8*(lane>=16), N = lane&15
    const int n = nt * 16 + l;
#pragma unroll
    for (int r = 0; r < 8; ++r) {
        const int m = mt * 16 + kh * 8 + r;
        C[(size_t)m * N + n] = acc[r];
    }
}

// ---------------------------------------------------------------------------
// Fused per-(batch,head) state update. One 128-thread block per (b,h) pair.
// Hot loop is b128-vectorized: 4 state rows x 32 column-quads per iteration,
// with non-temporal load/store on the stream-once h / h_new tensors so the
// 268 MB stream does not churn the 192 MB L2.
// ---------------------------------------------------------------------------
__global__ __launch_bounds__(128) void mamba_state_update(
    const float* __restrict__ proj,
    const float* __restrict__ h,
    const float* __restrict__ B_prev,
    const float* __restrict__ x_prev,
    const float* __restrict__ theta_cum,
    const float* __restrict__ A_log,
    const float* __restrict__ B_bias,
    const float* __restrict__ C_bias,
    float* __restrict__ out,            // d_out base
    float* __restrict__ g)              // gated y (BATCH, D_INNER) scratch
{
    const int b  = blockIdx.x >> 5;     // / N_HEADS
    const int hd = blockIdx.x & 31;     // % N_HEADS
    const int t  = threadIdx.x;         // 0..127

    __shared__ float sB[D_STATE], sC[D_STATE], sBp[D_STATE], red[128];
    __shared__ float sYp[4 * HEAD_DIM]; // partial y per row-group

    const float* prow = proj + (size_t)b * PROJ_DIM;
    const float braw = prow[BOFF + hd * D_STATE + t];
    const float craw = prow[COFF + hd * D_STATE + t];
    const float ea   = __expf(A_log[hd * D_STATE + t]);

    auto block_sum = [&](float v) -> float {
        red[t] = v; __syncthreads();
        for (int s = 64; s > 0; s >>= 1) {
            if (t < s) red[t] += red[t + s];
            __syncthreads();
        }
        float r = red[0]; __syncthreads();
        return r;
    };

    const float sb2 = block_sum(braw * braw);
    const float sc2 = block_sum(craw * craw);
    const float sea = block_sum(ea);

    const float rinvB = rsqrtf(sb2 * (1.0f / D_STATE) + 1e-6f);
    const float rinvC = rsqrtf(sc2 * (1.0f / D_STATE) + 1e-6f);
    const float Acoef = -sea * (1.0f / D_STATE);

    sB[t]  = braw * rinvB + B_bias[hd * D_STATE + t];
    sC[t]  = craw * rinvC + C_bias[hd * D_STATE + t];
    sBp[t] = B_prev[((size_t)b * N_HEADS + hd) * D_STATE + t];

    const float dtr  = prow[DTOFF + hd];
    const float lamr = prow[LAMOFF + hd];
    const float dt   = (dtr > 20.0f) ? dtr : log1pf(__expf(dtr));
    const float lam  = 1.0f / (1.0f + __expf(-lamr));
    const float alpha = __expf(dt * Acoef);
    const float beta  = (1.0f - lam) * dt * alpha;
    const float gamma = lam * dt;
    __syncthreads();

    // RoPE on B and C with updated theta cumsum (thread t owns pair t)
    if (t < D_STATE / 2) {
        const size_t thix = ((size_t)b * N_HEADS + hd) * (D_STATE / 2) + t;
        const float th = theta_cum[thix] + prow[THOFF + hd * (D_STATE / 2) + t] * dt;
        out[OFF_TH + thix] = th;
        const float c = __cosf(th), s = __sinf(th);
        const float b0 = sB[2 * t], b1 = sB[2 * t + 1];
        sB[2 * t]     = b0 * c - b1 * s;
        sB[2 * t + 1] = b0 * s + b1 * c;
        const float c0 = sC[2 * t], c1 = sC[2 * t + 1];
        sC[2 * t]     = c0 * c - c1 * s;
        sC[2 * t + 1] = c0 * s + c1 * c;
    }
    __syncthreads();

    // --- vectorized h update: thread owns columns c4..c4+3 of row-group grp
    const int grp = t >> 5;             // 0..3 (row subset: n = 4*i + grp)
    const int c4  = (t & 31) * 4;       // column base
    const size_t base128 = ((size_t)b * N_HEADS + hd) * HEAD_DIM;

    const v4f xp4  = *(const v4f*)(prow + XOFF + hd * HEAD_DIM + c4);
    const v4f xpv4 = *(const v4f*)(x_prev + base128 + c4);

    const size_t hbase = ((size_t)b * N_HEADS + hd) * (size_t)(D_STATE * HEAD_DIM);
    const v4f* hin  = (const v4f*)(h + hbase);
    v4f*       hout = (v4f*)(out + OFF_H + hbase);
    const int  q    = t & 31;           // quad index within a row (0..31)

    v4f y4 = {0.0f, 0.0f, 0.0f, 0.0f};
#pragma unroll 4
    for (int n0 = 0; n0 < D_STATE; n0 += 4) {
        const int n  = n0 + grp;
        const float cb = beta  * sBp[n];
        const float cg = gamma * sB[n];
        const float cc = sC[n];
        const v4f hv = __builtin_nontemporal_load(hin + (size_t)n * (HEAD_DIM / 4) + q);
        v4f hn;
        hn.x = fmaf(alpha, hv.x, fmaf(cb, xpv4.x, cg * xp4.x));
        hn.y = fmaf(alpha, hv.y, fmaf(cb, xpv4.y, cg * xp4.y));
        hn.z = fmaf(alpha, hv.z, fmaf(cb, xpv4.z, cg * xp4.z));
        hn.w = fmaf(alpha, hv.w, fmaf(cb, xpv4.w, cg * xp4.w));
        __builtin_nontemporal_store(hn, hout + (size_t)n * (HEAD_DIM / 4) + q);
        y4.x = fmaf(cc, hn.x, y4.x);
        y4.y = fmaf(cc, hn.y, y4.y);
        y4.z = fmaf(cc, hn.z, y4.z);
        y4.w = fmaf(cc, hn.w, y4.w);
    }

    // reduce the 4 row-group partials down to one y per column
    sYp[grp * HEAD_DIM + c4 + 0] = y4.x;
    sYp[grp * HEAD_DIM + c4 + 1] = y4.y;
    sYp[grp * HEAD_DIM + c4 + 2] = y4.z;
    sYp[grp * HEAD_DIM + c4 + 3] = y4.w;
    __syncthreads();

    const float y = sYp[0 * HEAD_DIM + t] + sYp[1 * HEAD_DIM + t] +
                    sYp[2 * HEAD_DIM + t] + sYp[3 * HEAD_DIM + t];

    const float zp   = prow[ZOFF + hd * HEAD_DIM + t];
    const float silu = zp / (1.0f + __expf(-zp));
    g[(size_t)b * D_INNER + hd * HEAD_DIM + t] = y * silu;

    out[OFF_B  + ((size_t)b * N_HEADS + hd) * D_STATE + t] = sB[t];
    out[OFF_XP + base128 + t] = prow[XOFF + hd * HEAD_DIM + t];
}

extern "C" void kernel_launch(void* const* d_in, const int* in_sizes, int n_in,
                              void* d_out, int out_size, void* d_ws, size_t ws_size,
                              hipStream_t stream) {
    const float* x      = (const float*)d_in[0];
    const float* h      = (const float*)d_in[1];
    const float* B_prev = (const float*)d_in[2];
    const float* x_prev = (const float*)d_in[3];
    const float* th_cum = (const float*)d_in[4];
    const float* A_log  = (const float*)d_in[5 + 1]; // placeholder fixed below
    (void)A_log;
    A_log               = (const float*)d_in[6];
    const float* W_in   = (const float*)d_in[5];
    const float* B_bias = (const float*)d_in[7];
    const float* C_bias = (const float*)d_in[8];
    const float* W_out  = (const float*)d_in[9];

    float* out  = (float*)d_out;
    float* proj = (float*)d_ws;                       // BATCH x PROJ_DIM
    float* g    = proj + (size_t)BATCH * PROJ_DIM;    // BATCH x D_INNER

    // Phase 1: proj = x @ W_in^T   (M=64, N=18496, K=2048)
    {
        const int tiles  = (BATCH / 16) * (PROJ_DIM / 16);  // 4624 wave-tiles
        const int blocks = (tiles + 7) / 8;                 // 8 waves / 256-thr block
        gemm_abt_wmma_f32<<<blocks, 256, 0, stream>>>(x, W_in, proj,
                                                      BATCH, PROJ_DIM, D_MODEL);
    }

    // Phase 2: fused state update (2048 blocks x 128 threads)
    mamba_state_update<<<BATCH * N_HEADS, 128, 0, stream>>>(
        proj, h, B_prev, x_prev, th_cum, A_log, B_bias, C_bias, out, g);

    // Phase 3: y = g @ W_out^T   (M=64, N=2048, K=4096) -> d_out[OFF_Y]
    {
        const int tiles  = (BATCH / 16) * (D_MODEL / 16);   // 512 wave-tiles
        const int blocks = (tiles + 7) / 8;
        gemm_abt_wmma_f32<<<blocks, 256, 0, stream>>>(g, W_out, out + OFF_Y,
                                                      BATCH, D_MODEL, D_INNER);
    }
}